// Mamba_v1_12240656793950
// MI455X (gfx1250) — compile-verified
//
#include <hip/hip_runtime.h>
#include <math.h>

// ---------------------------------------------------------------------------
// MI455X (gfx1250) implementation. All heavy GEMMs use V_WMMA_F32_16X16X4_F32
// (wave32). Pipeline is HBM-bound (~1.6 GB traffic => ~70us at 23.3 TB/s);
// fp32 WMMA keeps full precision while easily covering the ~115 GFLOP.
// GEMMs are LDS-tiled (32x32 per 4-wave block, padded stride 36 => bank-
// conflict-free fragment reads); conv3x3 uses tap-major implicit GEMM with
// branch-free zero padding (no div/mod, no exec divergence in inner loop).
// ---------------------------------------------------------------------------

typedef __attribute__((ext_vector_type(2))) float v2f;
typedef __attribute__((ext_vector_type(8))) float v8f;

#define NB   8
#define NC   512
#define HW   16384   // 128*128
#define LSEQ 256
#define DMOD 1024
#define DINN 2048
#define NST  16
#define DTRK 64
#define XDBW 96      // dt_rank + 2*d_state

// ---------------- helpers ----------------
__device__ __forceinline__ float cubic_w_(float x) {
  const float a = -0.75f;
  x = fabsf(x);
  if (x <= 1.f) return ((a + 2.f) * x - (a + 3.f)) * x * x + 1.f;
  if (x < 2.f)  return (((x - 5.f) * x + 8.f) * x - 4.f) * a;
  return 0.f;
}

__device__ __forceinline__ void bicubic_taps_(int o, float scale, int n_in,
                                              float* w, int* idx) {
  float src = ((float)o + 0.5f) * scale - 0.5f;
  float fi  = floorf(src);
  int   i0  = (int)fi;
  float t   = src - fi;
#pragma unroll
  for (int k = -1; k <= 2; ++k) {
    int i = i0 + k;
    i = i < 0 ? 0 : (i > n_in - 1 ? n_in - 1 : i);
    idx[k + 1] = i;
    w[k + 1]   = cubic_w_((float)k - t);
  }
}

__device__ __forceinline__ float silu_(float x) { return x / (1.f + expf(-x)); }
__device__ __forceinline__ float softplus_(float x) {
  return x > 20.f ? x : log1pf(expf(x));
}

// ---------------- per-(b,c) mean / std (ddof=1, eps=1e-5) ----------------
__global__ void stats_kernel(const float* __restrict__ x, int n,
                             float* __restrict__ mean, float* __restrict__ stdv) {
  int g = blockIdx.x;
  const float* p = x + (size_t)g * n;
  float s = 0.f, s2 = 0.f;
  for (int i = threadIdx.x; i < n; i += 256) {
    float v = p[i];
    s += v; s2 += v * v;
  }
  __shared__ float sh[512];
  sh[threadIdx.x] = s;
  sh[256 + threadIdx.x] = s2;
  __syncthreads();
  for (int o = 128; o > 0; o >>= 1) {
    if (threadIdx.x < o) {
      sh[threadIdx.x]       += sh[threadIdx.x + o];
      sh[256 + threadIdx.x] += sh[256 + threadIdx.x + o];
    }
    __syncthreads();
  }
  if (threadIdx.x == 0) {
    float m   = sh[0] / (float)n;
    float var = (sh[256] - (float)n * m * m) / (float)(n - 1);
    mean[g] = m;
    stdv[g] = sqrtf(var + 1e-5f);
  }
}

// ---------------- 3x3 conv (16x16 spatial) as implicit GEMM, WMMA ----------
// Tap-major K order: k = tap*Cin + ic. Per-tap geometry hoisted out of the
// inner ic loop; zero padding via clamped address * 0/1 mask (branch-free).
// x:[B,Cin,16,16]  w:[Cout,Cin,3,3]  out:[B,Cout,16,16]. Wave per 16oc x 16px.
__global__ void conv3x3_wmma(const float* __restrict__ x, const float* __restrict__ w,
                             const float* __restrict__ bias, float* __restrict__ out,
                             int cin, int cout, int do_relu) {
  int m0   = blockIdx.x * 16;          // oc tile
  int tile = blockIdx.y;               // pixel tile (16 tiles of 16)
  int b    = blockIdx.z;
  int lane = threadIdx.x;
  int r    = lane & 15;
  int kks  = (lane >> 4) * 2;
  int pix  = tile * 16 + r;
  int py   = pix >> 4, px = pix & 15;
  int K9   = cin * 9;
  const float* xb = x + (size_t)b * cin * 256;

  v8f acc = {};
#pragma unroll
  for (int tap = 0; tap < 9; ++tap) {
    int iy = py + tap / 3 - 1;
    int ix = px + tap % 3 - 1;
    float maskf = (iy >= 0 && iy < 16 && ix >= 0 && ix < 16) ? 1.f : 0.f;
    int cy = iy < 0 ? 0 : (iy > 15 ? 15 : iy);
    int cx = ix < 0 ? 0 : (ix > 15 ? 15 : ix);
    int off = cy * 16 + cx;
    const float* wb = w + (size_t)(m0 + r) * K9 + tap;
    for (int ic = 0; ic < cin; ic += 4) {
      v2f a, bf;
      a[0]  = wb[(ic + kks) * 9];
      a[1]  = wb[(ic + kks + 1) * 9];
      bf[0] = xb[(ic + kks) * 256 + off] * maskf;
      bf[1] = xb[(ic + kks + 1) * 256 + off] * maskf;
      acc = __builtin_amdgcn_wmma_f32_16x16x4_f32(false, a, false, bf,
                                                  (short)0, acc, false, false);
    }
  }
  int mo = (lane >> 4) * 8;
#pragma unroll
  for (int v = 0; v < 8; ++v) {
    int oc = m0 + v + mo;
    float val = acc[v] + bias[oc];
    if (do_relu) val = fmaxf(val, 0.f);
    out[((size_t)b * cout + oc) * 256 + tile * 16 + (lane & 15)] = val;
  }
}

// ---------------- LDS-tiled GEMM: C[m,n] = sum_k A[m,k] * W[n,k] -----------
// 128 threads (4 waves) per block; block computes a 32x32 C tile; A/W k-chunks
// (32 wide) staged in LDS via float4 (ds_store_b128), row stride 36 DWORDs so
// fragment reads are bank-conflict-free and stores stay 16B aligned.
// Requires M%32==0, N%32==0, K%32==0 (true for all uses here).
#define GSTRIDE 36
__global__ void gemm_nk_wmma(const float* __restrict__ A, int lda,
                             const float* __restrict__ W, int ldw,
                             float* __restrict__ C, int ldc, int K) {
  int m0  = blockIdx.x * 32, n0 = blockIdx.y * 32;
  int tid = threadIdx.x;
  int wave = tid >> 5, lane = tid & 31;
  int r   = lane & 15;
  int kks = (lane >> 4) * 2;
  int mt  = (wave >> 1) * 16;   // wave's 16-row offset inside 32x32 tile
  int nt  = (wave & 1) * 16;    // wave's 16-col offset

  __shared__ float As[32 * GSTRIDE];
  __shared__ float Ws[32 * GSTRIDE];

  v8f acc = {};
  for (int k = 0; k < K; k += 32) {
    __syncthreads();   // previous chunk's fragment reads complete
    // 32 rows x 32 floats = 256 float4 per matrix; 2 per thread
    for (int f = tid; f < 256; f += 128) {
      int row = f >> 3, c4 = (f & 7) * 4;
      *(float4*)&As[row * GSTRIDE + c4] =
          *(const float4*)(A + (size_t)(m0 + row) * lda + k + c4);
      *(float4*)&Ws[row * GSTRIDE + c4] =
          *(const float4*)(W + (size_t)(n0 + row) * ldw + k + c4);
    }
    __syncthreads();
#pragma unroll
    for (int kk = 0; kk < 32; kk += 4) {
      v2f a, bf;
      a[0]  = As[(mt + r) * GSTRIDE + kk + kks];
      a[1]  = As[(mt + r) * GSTRIDE + kk + kks + 1];
      bf[0] = Ws[(nt + r) * GSTRIDE + kk + kks];
      bf[1] = Ws[(nt + r) * GSTRIDE + kk + kks + 1];
      acc = __builtin_amdgcn_wmma_f32_16x16x4_f32(false, a, false, bf,
                                                  (short)0, acc, false, false);
    }
  }
  int mo = (lane >> 4) * 8;
#pragma unroll
  for (int v = 0; v < 8; ++v)
    C[(size_t)(m0 + mt + v + mo) * ldc + n0 + nt + r] = acc[v];
}

// ---------------- build mamba input sequence [b, pix, 1024] -----------------
// ch<512: bicubic-128->16 of F_content, then (x-m_c)/s_c  (mvn commutes w/ resize)
// ch>=512: (fs - m_s)/s_s
__global__ void combine_kernel(const float* __restrict__ Fc, const float* __restrict__ fs,
                               const float* __restrict__ m_c, const float* __restrict__ s_c,
                               const float* __restrict__ m_s, const float* __restrict__ s_s,
                               float* __restrict__ seq) {
  int idx = blockIdx.x * 256 + threadIdx.x;           // 8*256*1024
  int ch  = idx & 1023;
  int pix = (idx >> 10) & 255;
  int b   = idx >> 18;
  float v;
  if (ch < NC) {
    int c = ch;
    int oy = pix >> 4, ox = pix & 15;
    float wy[4], wx[4];
    int   iy[4], ix[4];
    bicubic_taps_(oy, 8.f, 128, wy, iy);
    bicubic_taps_(ox, 8.f, 128, wx, ix);
    const float* base = Fc + ((size_t)b * NC + c) * HW;
    float acc = 0.f;
#pragma unroll
    for (int u = 0; u < 4; ++u)
#pragma unroll
      for (int w = 0; w < 4; ++w)
        acc += wy[u] * wx[w] * base[iy[u] * 128 + ix[w]];
    int g = b * NC + c;
    v = (acc - m_c[g]) / s_c[g];
  } else {
    int c = ch - NC;
    int g = b * NC + c;
    v = (fs[((size_t)b * NC + c) * 256 + pix] - m_s[g]) / s_s[g];
  }
  seq[idx] = v;
}

// ---------------- causal depthwise conv1d (d_conv=4) + SiLU ----------------
__global__ void conv1d_silu_kernel(const float* __restrict__ xz,
                                   const float* __restrict__ w,
                                   const float* __restrict__ bias,
                                   float* __restrict__ xc) {
  int idx = blockIdx.x * 256 + threadIdx.x;           // 8*256*2048
  int d = idx & 2047;
  int l = (idx >> 11) & 255;
  int b = idx >> 19;
  float acc = bias[d];
#pragma unroll
  for (int j = 0; j < 4; ++j) {
    int ll = l - 3 + j;
    if (ll >= 0)
      acc += w[d * 4 + j] * xz[((size_t)b * LSEQ + ll) * (2 * DINN) + d];
  }
  xc[idx] = silu_(acc);
}

// ---------------- dt = softplus(gemm + bias), in place ----------------------
__global__ void softplus_bias_kernel(float* __restrict__ dt,
                                     const float* __restrict__ bias) {
  int idx = blockIdx.x * 256 + threadIdx.x;           // 8*256*2048
  int d = idx & 2047;
  dt[idx] = softplus_(dt[idx] + bias[d]);
}

// ---------------- selective scan + gating ----------------------------------
// one thread per (b,d); B/C of the current timestep staged in LDS.
__global__ void scan_kernel(const float* __restrict__ xdb, const float* __restrict__ dt,
                            const float* __restrict__ xc, const float* __restrict__ xz,
                            const float* __restrict__ A_log, const float* __restrict__ Dp,
                            float* __restrict__ y) {
  int b = blockIdx.y;
  int d = blockIdx.x * 256 + threadIdx.x;
  float Ar[NST], h[NST];
#pragma unroll
  for (int n = 0; n < NST; ++n) {
    Ar[n] = -expf(A_log[d * NST + n]);
    h[n]  = 0.f;
  }
  float Dv = Dp[d];
  __shared__ float sBC[32];
  for (int l = 0; l < LSEQ; ++l) {
    if (threadIdx.x < 32)
      sBC[threadIdx.x] = xdb[((size_t)b * LSEQ + l) * XDBW + DTRK + threadIdx.x];
    __syncthreads();
    size_t row = (size_t)b * LSEQ + l;
    float dtv = dt[row * DINN + d];
    float xcv = xc[row * DINN + d];
    float acc = 0.f;
#pragma unroll
    for (int n = 0; n < NST; ++n) {
      float dA = expf(dtv * Ar[n]);
      h[n] = dA * h[n] + dtv * sBC[n] * xcv;
      acc += h[n] * sBC[16 + n];
    }
    float zv = xz[row * (2 * DINN) + DINN + d];
    y[row * DINN + d] = (acc + xcv * Dv) * silu_(zv);
    __syncthreads();
  }
}

// ---------------- bicubic 16->128 upsample of res_seq(:, :512) + F_c residual
// writes into d_out, which doubles as the 268MB scratch for the final stage.
__global__ void upsample_add_kernel(const float* __restrict__ res_seq,
                                    const float* __restrict__ Fc,
                                    const float* __restrict__ m_c,
                                    const float* __restrict__ s_c,
                                    float* __restrict__ outp) {
  size_t idx = (size_t)blockIdx.x * 256 + threadIdx.x; // 8*512*16384
  int pix = (int)(idx & 16383);
  int c   = (int)((idx >> 14) & 511);
  int b   = (int)(idx >> 23);
  int oy = pix >> 7, ox = pix & 127;
  float wy[4], wx[4];
  int   iy[4], ix[4];
  bicubic_taps_(oy, 0.125f, 16, wy, iy);
  bicubic_taps_(ox, 0.125f, 16, wx, ix);
  float acc = 0.f;
#pragma unroll
  for (int u = 0; u < 4; ++u)
#pragma unroll
    for (int w = 0; w < 4; ++w)
      acc += wy[u] * wx[w] *
             res_seq[((size_t)b * LSEQ + (iy[u] * 16 + ix[w])) * DMOD + c];
  int g = b * NC + c;
  float fc = (Fc[idx] - m_c[g]) / s_c[g];
  outp[idx] = acc + fc;
}

// ---------------- final 1x1 conv, in place on d_out -------------------------
// input renormalized:  R = (d_out - m_r)/s_r * s_s + m_s
// block = one batch, 16 pixels; R[512ic][16px] staged in 32KB LDS, then
// 8 waves x 4 oc-tiles of WMMA over K=512, writing d_out in place.
__global__ void final_conv_wmma(float* __restrict__ outp,
                                const float* __restrict__ w1x1,
                                const float* __restrict__ bias,
                                const float* __restrict__ m_r, const float* __restrict__ s_r,
                                const float* __restrict__ m_s, const float* __restrict__ s_s) {
  int tile = blockIdx.x;          // 1024 pixel tiles
  int b    = blockIdx.y;
  __shared__ float ls[NC * 16];   // 32 KB of the 320 KB WGP LDS
  for (int i = threadIdx.x; i < NC * 16; i += 256) {
    int ic = i >> 4, n = i & 15;
    int g  = b * NC + ic;
    float v = outp[((size_t)b * NC + ic) * HW + tile * 16 + n];
    ls[i] = (v - m_r[g]) / s_r[g] * s_s[g] + m_s[g];
  }
  __syncthreads();

  int wave = threadIdx.x >> 5;
  int lane = threadIdx.x & 31;
  int r   = lane & 15;
  int kks = (lane >> 4) * 2;
  int mo  = (lane >> 4) * 8;
  for (int t = 0; t < 4; ++t) {
    int oc0 = (wave * 4 + t) * 16;
    v8f acc = {};
    for (int k = 0; k < NC; k += 4) {
      v2f a, bf;
      const float* ap = w1x1 + (size_t)(oc0 + r) * NC + k + kks;
      a[0]  = ap[0];
      a[1]  = ap[1];
      bf[0] = ls[(k + kks) * 16 + r];
      bf[1] = ls[(k + kks + 1) * 16 + r];
      acc = __builtin_amdgcn_wmma_f32_16x16x4_f32(false, a, false, bf,
                                                  (short)0, acc, false, false);
    }
#pragma unroll
    for (int v = 0; v < 8; ++v) {
      int oc = oc0 + v + mo;
      outp[((size_t)b * NC + oc) * HW + tile * 16 + (lane & 15)] = acc[v] + bias[oc];
    }
  }
}

// ---------------------------------------------------------------------------
extern "C" void kernel_launch(void* const* d_in, const int* in_sizes, int n_in,
                              void* d_out, int out_size, void* d_ws, size_t ws_size,
                              hipStream_t stream) {
  const float* F_clip   = (const float*)d_in[0];
  const float* F_cont   = (const float*)d_in[1];
  const float* fs_w1    = (const float*)d_in[2];
  const float* fs_b1    = (const float*)d_in[3];
  const float* fs_w2    = (const float*)d_in[4];
  const float* fs_b2    = (const float*)d_in[5];
  const float* out_w    = (const float*)d_in[6];
  const float* out_b    = (const float*)d_in[7];
  const float* in_proj  = (const float*)d_in[8];
  const float* c1d_w    = (const float*)d_in[9];
  const float* c1d_b    = (const float*)d_in[10];
  const float* x_proj   = (const float*)d_in[11];
  const float* dt_proj  = (const float*)d_in[12];
  const float* dt_pb    = (const float*)d_in[13];
  const float* A_log    = (const float*)d_in[14];
  const float* D_p      = (const float*)d_in[15];
  const float* out_proj = (const float*)d_in[16];
  float* outp = (float*)d_out;

  // workspace layout (~110 MB of f32)
  float* Wk = (float*)d_ws;
  size_t o = 0;
  float* fs_tmp = Wk + o; o += (size_t)NB * NC * 256;
  float* fs     = Wk + o; o += (size_t)NB * NC * 256;
  float* m_c = Wk + o; o += NB * NC;
  float* s_c = Wk + o; o += NB * NC;
  float* m_s = Wk + o; o += NB * NC;
  float* s_s = Wk + o; o += NB * NC;
  float* m_r = Wk + o; o += NB * NC;
  float* s_r = Wk + o; o += NB * NC;
  float* seq = Wk + o; o += (size_t)NB * LSEQ * DMOD;
  float* xz  = Wk + o; o += (size_t)NB * LSEQ * 2 * DINN;
  float* xc  = Wk + o; o += (size_t)NB * LSEQ * DINN;
  float* xdb = Wk + o; o += (size_t)NB * LSEQ * XDBW;
  float* dt  = Wk + o; o += (size_t)NB * LSEQ * DINN;
  float* yb  = Wk + o; o += (size_t)NB * LSEQ * DINN;
  float* rsq = Wk + o; o += (size_t)NB * LSEQ * DMOD;
  (void)ws_size; (void)in_sizes; (void)n_in; (void)out_size;

  // 1. per-(b,c) stats of F_content
  stats_kernel<<<NB * NC, 256, 0, stream>>>(F_cont, HW, m_c, s_c);
  // 2-3. fs = conv2(relu(conv1(F_clip)))
  conv3x3_wmma<<<dim3(NC / 16, 16, NB), 32, 0, stream>>>(F_clip, fs_w1, fs_b1, fs_tmp, NC, NC, 1);
  conv3x3_wmma<<<dim3(NC / 16, 16, NB), 32, 0, stream>>>(fs_tmp, fs_w2, fs_b2, fs, NC, NC, 0);
  // 4. stats of fs
  stats_kernel<<<NB * NC, 256, 0, stream>>>(fs, 256, m_s, s_s);
  // 5. combine -> seq [b, 256, 1024]
  combine_kernel<<<(NB * LSEQ * DMOD) / 256, 256, 0, stream>>>(F_cont, fs, m_c, s_c, m_s, s_s, seq);
  // 6. in_proj: [2048 x 1024] @ [4096 x 1024]^T
  gemm_nk_wmma<<<dim3(NB * LSEQ / 32, 2 * DINN / 32), 128, 0, stream>>>(seq, DMOD, in_proj, DMOD, xz, 2 * DINN, DMOD);
  // 7. depthwise conv1d + silu
  conv1d_silu_kernel<<<(NB * LSEQ * DINN) / 256, 256, 0, stream>>>(xz, c1d_w, c1d_b, xc);
  // 8. x_proj: [2048 x 2048] @ [96 x 2048]^T
  gemm_nk_wmma<<<dim3(NB * LSEQ / 32, XDBW / 32), 128, 0, stream>>>(xc, DINN, x_proj, DINN, xdb, XDBW, DINN);
  // 9. dt gemm: [2048 x 64] @ [2048 x 64]^T
  gemm_nk_wmma<<<dim3(NB * LSEQ / 32, DINN / 32), 128, 0, stream>>>(xdb, XDBW, dt_proj, DTRK, dt, DINN, DTRK);
  // 10. dt = softplus(dt + b)
  softplus_bias_kernel<<<(NB * LSEQ * DINN) / 256, 256, 0, stream>>>(dt, dt_pb);
  // 11. selective scan + gating -> y
  scan_kernel<<<dim3(DINN / 256, NB), 256, 0, stream>>>(xdb, dt, xc, xz, A_log, D_p, yb);
  // 12. out_proj: [2048 x 2048] @ [1024 x 2048]^T
  gemm_nk_wmma<<<dim3(NB * LSEQ / 32, DMOD / 32), 128, 0, stream>>>(yb, DINN, out_proj, DINN, rsq, DMOD, DINN);
  // 13. upsample 16->128 + F_c residual -> d_out (scratch)
  upsample_add_kernel<<<(int)(((size_t)NB * NC * HW) / 256), 256, 0, stream>>>(rsq, F_cont, m_c, s_c, outp);
  // 14. stats of the residual image
  stats_kernel<<<NB * NC, 256, 0, stream>>>(outp, HW, m_r, s_r);
  // 15. renorm + 1x1 conv, in place on d_out
  final_conv_wmma<<<dim3(HW / 16, NB), 256, 0, stream>>>(outp, out_w, out_b, m_r, s_r, m_s, s_s);
}